// Attention_w_viz_48284022342315
// MI455X (gfx1250) — compile-verified
//
#include <hip/hip_runtime.h>
#include <math.h>

// ---------------------------------------------------------------------------
// s = einsum('sbh,h->sb', X, W) + b ; reverse cumulative logsumexp over S;
// out = exp(s - suffix_lse).  S=4096, B=64, H=512, f32.
//
// Kernel 1: TDM (tensor_load_to_lds) streams 128x64-f32 tiles HBM->LDS with
//   per-row padding (pitch 66 dwords, bank-conflict-free ds_load_b64), double
//   buffered against 8 waves doing chained v_wmma_f32_16x16x4_f32 mat-vec.
//   One wave = 16 rows; block = 8 waves = 128 rows; 2048 blocks.
// Kernel 2: per-column reverse logsumexp scan (L2-resident, negligible).
// ---------------------------------------------------------------------------

typedef __attribute__((ext_vector_type(2))) float        v2f;
typedef __attribute__((ext_vector_type(8))) float        v8f;
typedef __attribute__((ext_vector_type(4))) unsigned int v4u;
typedef __attribute__((ext_vector_type(8))) int          v8i;
typedef __attribute__((ext_vector_type(4))) int          v4i;

#define S_LEN   4096
#define B_COLS  64
#define H_DIM   512
#define KC      64            // K-chunk per tile (dwords)
#define ROWS_PB 128           // rows per block
#define PITCH   66            // KC + 2 pad dwords (TDM pad_interval=64, pad_amount=2)

__device__ __forceinline__ void tdm_load_tile(const float* gsrc, unsigned lds_byte_off)
{
    // --- D# group 0: count=1, lds_addr, global_addr, type=2 ---
    unsigned long long ga = (unsigned long long)(uintptr_t)gsrc;  // byte addr
    v4u g0;
    g0[0] = 1u;                                   // count=1, user mode
    g0[1] = lds_byte_off;                         // LDS byte address
    g0[2] = (unsigned)(ga & 0xFFFFFFFFull);       // global_addr[31:0]
    g0[3] = (unsigned)((ga >> 32) & 0x01FFFFFFull) | (2u << 30);  // [56:32] | type=2

    // --- D# group 1 ---
    // data_size=2 (4B), pad_enable=1, pad_interval=5 (64 dw), pad_amount=1 (2 dw)
    v8i g1;
    g1[0] = (int)((2u << 16) | (1u << 20) | (5u << 22) | (1u << 25));
    g1[1] = (int)(512u << 16);                    // tensor_dim0=512 -> bits[79:48]
    g1[2] = 0;                                    // dim0 hi16=0, tensor_dim1 lo16=0
    g1[3] = (int)(4u | (64u << 16));              // tensor_dim1=262144 hi16=4 | tile_dim0=64
    g1[4] = (int)(128u);                          // tile_dim1=128, tile_dim2=0
    g1[5] = (int)(512u);                          // tensor_dim0_stride=512 (lo32)
    g1[6] = 0;                                    // stride0 hi16=0, stride1 lo16=0
    g1[7] = 0;

    v4i gz4 = {0, 0, 0, 0};                       // groups 2/3 unused (2D tensor)
    v8i gz8 = {0, 0, 0, 0, 0, 0, 0, 0};
    __builtin_amdgcn_tensor_load_to_lds(g0, g1, gz4, gz4, gz8, 0);
}

__launch_bounds__(256, 2)
__global__ void fc_scores_wmma(const float* __restrict__ X,
                               const float* __restrict__ W,
                               const float* __restrict__ bias,
                               float* __restrict__ sc)   // sc[b][t], col-major
{
    __shared__ float lw[H_DIM];
    __shared__ float tiles[2][ROWS_PB * PITCH];   // 2 x 33792 B

    for (int i = threadIdx.x; i < H_DIM; i += 256) lw[i] = W[i];

    const int tid  = threadIdx.x;
    const int lane = tid & 31;
    const int wave = tid >> 5;
    const int m    = lane & 15;                   // A row within wave's 16
    const int half = lane >> 4;                   // 0: K=0,1  1: K=2,3
    const long long row0 = (long long)blockIdx.x * ROWS_PB;   // block's first row

    const float* gbase = X + row0 * H_DIM;        // tile origin advances by KC
    const unsigned ldsA = (unsigned)(uintptr_t)(void*)&tiles[0][0];
    const unsigned ldsB = (unsigned)(uintptr_t)(void*)&tiles[1][0];

    // prologue: DMA tile 0 into buffer 0
    if (wave == 0) {
        tdm_load_tile(gbase, ldsA);
        __builtin_amdgcn_s_wait_tensorcnt(0);
    }
    __syncthreads();

    v8f c = {0.f, 0.f, 0.f, 0.f, 0.f, 0.f, 0.f, 0.f};
    const int rowOff = (wave * 16 + m) * PITCH + 2 * half;   // dword offset in tile

    #pragma unroll
    for (int tile = 0; tile < H_DIM / KC; ++tile) {          // 8 tiles
        const int cur = tile & 1;
        // prefetch next tile into the other buffer
        if (wave == 0 && tile + 1 < H_DIM / KC) {
            tdm_load_tile(gbase + (long long)(tile + 1) * KC,
                          (cur ? ldsA : ldsB));
        }

        const float* bufc = &tiles[cur][0];
        const float* wp   = lw + tile * KC + 2 * half;
        #pragma unroll
        for (int kk = 0; kk < KC; kk += 4) {
            v2f a = *(const v2f*)(bufc + rowOff + kk);       // ds_load_b64, no conflicts
            v2f bfrag;
            bfrag.x = wp[kk];
            bfrag.y = wp[kk + 1];
            c = __builtin_amdgcn_wmma_f32_16x16x4_f32(
                    false, a, false, bfrag, (short)0, c, false, false);
        }

        if (wave == 0 && tile + 1 < H_DIM / KC) {
            __builtin_amdgcn_s_wait_tensorcnt(0);
        }
        __syncthreads();
    }

    const float bb = bias[0];
    // C layout: VGPR j, lanes 0-15 -> M=j; lanes 16-31 -> M=j+8 (columns identical)
    if (lane == 0 || lane == 16) {
        const long long base = row0 + wave * 16 + half * 8;
        #pragma unroll
        for (int j = 0; j < 8; ++j) {
            long long r = base + j;
            int t  = (int)(r >> 6);
            int bc = (int)(r & 63);
            sc[(long long)bc * S_LEN + t] = c[j] + bb;
        }
    }
}

__launch_bounds__(256)
__global__ void suffix_softmax(const float* __restrict__ sc,
                               float* __restrict__ out)  // out[t][b]
{
    __shared__ float sm[256];
    __shared__ float sz[256];
    const int bcol = blockIdx.x;       // 0..63
    const int tid  = threadIdx.x;      // 256 threads, 16 elems each
    const float* col = sc + (long long)bcol * S_LEN;
    const int t0 = tid * 16;

    float xv[16];
    #pragma unroll
    for (int j = 0; j < 16; ++j) xv[j] = col[t0 + j];

    float mC = -INFINITY, zC = 0.0f;
    #pragma unroll
    for (int j = 15; j >= 0; --j) {
        float x  = xv[j];
        float mn = fmaxf(mC, x);
        zC = zC * __expf(mC - mn) + __expf(x - mn);
        mC = mn;
    }
    sm[tid] = mC; sz[tid] = zC;
    __syncthreads();

    for (int off = 1; off < 256; off <<= 1) {
        const bool has = (tid + off) < 256;
        float om = has ? sm[tid + off] : -INFINITY;
        float oz = has ? sz[tid + off] : 0.0f;
        __syncthreads();
        float mn = fmaxf(mC, om);
        zC = zC * __expf(mC - mn) + oz * __expf(om - mn);
        mC = mn;
        sm[tid] = mC; sz[tid] = zC;
        __syncthreads();
    }

    float em = -INFINITY, ez = 0.0f;
    if (tid < 255) { em = sm[tid + 1]; ez = sz[tid + 1]; }

    float cm = em, cz = ez;
    #pragma unroll
    for (int j = 15; j >= 0; --j) {
        float x  = xv[j];
        float mn = fmaxf(cm, x);
        cz = cz * __expf(cm - mn) + __expf(x - mn);
        cm = mn;
        float lse = cm + __logf(cz);
        out[(long long)(t0 + j) * B_COLS + bcol] = __expf(x - lse);
    }
}

extern "C" void kernel_launch(void* const* d_in, const int* in_sizes, int n_in,
                              void* d_out, int out_size, void* d_ws, size_t ws_size,
                              hipStream_t stream) {
    (void)in_sizes; (void)n_in; (void)out_size; (void)ws_size;
    const float* X    = (const float*)d_in[0];   // [S, B, H] f32
    const float* W    = (const float*)d_in[1];   // [H, 1]   f32
    const float* bias = (const float*)d_in[2];   // [1]      f32
    float* out = (float*)d_out;                  // [S, B, 1] f32
    float* sc  = (float*)d_ws;                   // [B][S] scores, 1 MB

    fc_scores_wmma<<<2048, 256, 0, stream>>>(X, W, bias, sc);
    suffix_softmax<<<B_COLS, 256, 0, stream>>>(sc, out);
}